// MaxBeforePool1d_67817533603999
// MI455X (gfx1250) — compile-verified
//
#include <hip/hip_runtime.h>
#include <hip/hip_bf16.h>

// ---------------------------------------------------------------------------
// Causal sliding-window max:  out[b,c,l] = max(x[b,c, l-24 .. l]), pad=-inf
// Shapes: (64, 2048, 336) f32.  Memory bound: ~352 MB => ~15 us round trip at
// 23.3 TB/s.  One wave32 per row; async global->LDS staging (CDNA5 path);
// van Herk prefix/suffix sliding max in registers (~3.8 v_max per output);
// 12 outputs per lane on 28 lanes -> 3x global_store_b128 per lane.
// ---------------------------------------------------------------------------

namespace {
constexpr int kSeqL          = 336;          // row length
constexpr int kWin           = 25;           // window (kernel 24 + 1)
constexpr int kRowsTotal     = 64 * 2048;    // 131072 rows
constexpr int kRowsPerBlock  = 8;            // one wave32 per row
constexpr int kThreads       = 256;          // 8 waves
constexpr int kPad           = kWin - 1;     // 24-word front halo of -inf
constexpr int kLdsRow        = 384;          // 24 pad + 336 data (+24 slack)
constexpr int kChunksPerRow  = kSeqL / 4;    // 84 float4 chunks (16B aligned)
constexpr int kChunksPerBlk  = kRowsPerBlock * kChunksPerRow;  // 672
constexpr int kOutPerLane    = 12;           // 28 lanes * 12 = 336 exactly
constexpr int kActiveLanes   = kSeqL / kOutPerLane;            // 28
}

#if defined(__gfx1250__) && \
    __has_builtin(__builtin_amdgcn_global_load_async_to_lds_b128) && \
    __has_builtin(__builtin_amdgcn_s_wait_asynccnt)
#define USE_ASYNC_LDS 1
#else
#define USE_ASYNC_LDS 0
#endif

typedef int v4i __attribute__((ext_vector_type(4)));
typedef __attribute__((address_space(1))) v4i* global_v4i_p;
typedef __attribute__((address_space(3))) v4i* lds_v4i_p;

__global__ __launch_bounds__(kThreads)
void maxpool_causal_rows(const float* __restrict__ x, float* __restrict__ out) {
  __shared__ float tile[kRowsPerBlock * kLdsRow];

  const int tid  = threadIdx.x;
  const long long row0 = (long long)blockIdx.x * kRowsPerBlock;

  // ---- fill front halo with -inf (24 words per row, 192 total) ------------
  if (tid < kRowsPerBlock * kPad) {
    int w = tid / kPad;
    int q = tid % kPad;
    tile[w * kLdsRow + q] = -__builtin_inff();
  }

  // ---- stage 8 rows into LDS: async B128 copies (3 rounds x 256 thr) ------
#pragma unroll
  for (int it = 0; it < 3; ++it) {
    int ck = it * kThreads + tid;
    if (ck < kChunksPerBlk) {
      int w = ck / kChunksPerRow;
      int c = ck % kChunksPerRow;
      const float* g = x + (row0 + w) * kSeqL + 4 * c;       // 16B aligned
      float*       l = &tile[w * kLdsRow + kPad + 4 * c];    // 16B aligned
#if USE_ASYNC_LDS
      __builtin_amdgcn_global_load_async_to_lds_b128(
          (global_v4i_p)(void*)g,   // AS1 int4*  (cast drops const)
          (lds_v4i_p)(void*)l,      // AS3 int4*
          /*offset=*/0, /*cpol=*/0);
#else
      *(float4*)l = *(const float4*)g;
#endif
    }
  }
#if USE_ASYNC_LDS
  __builtin_amdgcn_s_wait_asynccnt(0);
#endif
  __syncthreads();

  // ---- compute: one wave per row, 12 outputs on each of 28 lanes ----------
  const int wv   = tid >> 5;
  const int lane = tid & 31;

  if (lane < kActiveLanes) {
    const int base = wv * kLdsRow + kOutPerLane * lane;   // max 3047 < 3072

    // a[k] = padded row word (base + k); window for output j is a[j..j+24]
    float a[kOutPerLane + kWin - 1];   // 36 values
#pragma unroll
    for (int k = 0; k < kOutPerLane + kWin - 1; ++k) a[k] = tile[base + k];

    // van Herk: s[j] = max(a[j..23]) (suffix), p_j = max(a[24..24+j]) (prefix)
    float t = a[23];
#pragma unroll
    for (int k = 22; k >= kOutPerLane; --k) t = fmaxf(t, a[k]);  // a[12..23]
    float s[kOutPerLane];
#pragma unroll
    for (int j = kOutPerLane - 1; j >= 0; --j) { t = fmaxf(t, a[j]); s[j] = t; }

    float o[kOutPerLane];
    float p = a[kPad];                 // a[24]
    o[0] = fmaxf(s[0], p);
#pragma unroll
    for (int j = 1; j < kOutPerLane; ++j) {
      p = fmaxf(p, a[kPad + j]);
      o[j] = fmaxf(s[j], p);
    }

    // ---- store: 3x b128, lane base stride 48B (16B aligned) ---------------
    float4* op = (float4*)(out + (row0 + wv) * kSeqL + kOutPerLane * lane);
    op[0] = make_float4(o[0], o[1], o[2],  o[3]);
    op[1] = make_float4(o[4], o[5], o[6],  o[7]);
    op[2] = make_float4(o[8], o[9], o[10], o[11]);
  }
}

extern "C" void kernel_launch(void* const* d_in, const int* in_sizes, int n_in,
                              void* d_out, int out_size, void* d_ws, size_t ws_size,
                              hipStream_t stream) {
  (void)in_sizes; (void)n_in; (void)out_size; (void)d_ws; (void)ws_size;
  const float* x = (const float*)d_in[0];
  float* out = (float*)d_out;
  dim3 grid(kRowsTotal / kRowsPerBlock);   // 16384 blocks
  dim3 block(kThreads);                    // 8 wave32 per block
  maxpool_causal_rows<<<grid, block, 0, stream>>>(x, out);
}